// AdaptiveEnhancementGate_27015344291949
// MI455X (gfx1250) — compile-verified
//
#include <hip/hip_runtime.h>
#include <hip/hip_bf16.h>

typedef __attribute__((ext_vector_type(2))) float v2f;
typedef __attribute__((ext_vector_type(8))) float v8f;

// ---------------------------------------------------------------------------
// Kernel 1: init workspace (counts, relation histogram, slot map)
// ---------------------------------------------------------------------------
__global__ void k_init(float* __restrict__ cnt, long long cntN,
                       float* __restrict__ relc, int R,
                       int* __restrict__ slot, const int* __restrict__ d_nn) {
    long long N = (long long)(*d_nn);
    long long total = cntN;
    if (N > total) total = N;
    if ((long long)R > total) total = R;
    long long stride = (long long)gridDim.x * blockDim.x;
    for (long long i = (long long)blockIdx.x * blockDim.x + threadIdx.x;
         i < total; i += stride) {
        if (i < cntN) cnt[i] = 0.0f;
        if (i < (long long)R) relc[i] = 0.0f;
        if (i < N) slot[i] = -1;
    }
}

// ---------------------------------------------------------------------------
// Kernel 2: mark query entities in the slot map (duplicates share a winner row)
// ---------------------------------------------------------------------------
__global__ void k_mark(const int* __restrict__ qe, int B,
                       const int* __restrict__ d_nn, int* __restrict__ slot) {
    int b = blockIdx.x * blockDim.x + threadIdx.x;
    if (b >= B) return;
    int N = *d_nn;
    int e = qe[b];
    if (e >= 0 && e < N) slot[e] = b;   // race benign: any winner's row holds the entity counts
}

// ---------------------------------------------------------------------------
// Kernel 3: edge scan -> per-(query-entity, relation) counts + relation histogram
// ---------------------------------------------------------------------------
__global__ void k_edges(const int* __restrict__ src, const int* __restrict__ dst,
                        const int* __restrict__ typ, long long E, int R,
                        const int* __restrict__ slot, const int* __restrict__ d_nn,
                        float* __restrict__ cnt, float* __restrict__ relc) {
    __shared__ unsigned hist[512];
    const bool useH = (R <= 512);
    int N = *d_nn;
    if (useH)
        for (int i = threadIdx.x; i < 512; i += blockDim.x) hist[i] = 0u;
    __syncthreads();

    long long stride = (long long)gridDim.x * blockDim.x;
    for (long long e = (long long)blockIdx.x * blockDim.x + threadIdx.x;
         e < E; e += stride) {
        int s = src[e], d = dst[e], t = typ[e];
        if (t < 0 || t >= R) continue;
        if (useH) atomicAdd(&hist[t], 1u);
        else      atomicAdd(&relc[t], 1.0f);
        if (s >= 0 && s < N) {
            int ss = slot[s];
            if (ss >= 0) atomicAdd(&cnt[(long long)ss * R + t], 1.0f);
        }
        if (d != s && d >= 0 && d < N) {
            int sd = slot[d];
            if (sd >= 0) atomicAdd(&cnt[(long long)sd * R + t], 1.0f);
        }
    }
    __syncthreads();
    if (useH)
        for (int i = threadIdx.x; i < R; i += blockDim.x) {
            unsigned h = hist[i];
            if (h) atomicAdd(&relc[i], (float)h);
        }
}

// ---------------------------------------------------------------------------
// Kernel 4: per-query feature build. One block (256 thr) per query b.
//   streams rel_embs[b] (134 MB total across grid) -> memory-bound part.
// ---------------------------------------------------------------------------
__global__ void k_feat(const float* __restrict__ emb, const int* __restrict__ qr,
                       const int* __restrict__ qe, const int* __restrict__ d_nn,
                       long long E, int R, int D, const int* __restrict__ slot,
                       const float* __restrict__ cnt, const float* __restrict__ relc,
                       float* __restrict__ feat, int KF) {
    __shared__ float s_cnt[512];       // R <= 512
    __shared__ float s_part[256];
    __shared__ float s_deg;
    __shared__ int s_slot, s_ent, s_qr, s_N;

    const int b = blockIdx.x;
    const int t = threadIdx.x;

    if (t == 0) {
        s_N = *d_nn;
        s_ent = qe[b];
        s_qr = qr[b];
        int sl = -1;
        if (s_ent >= 0 && s_ent < s_N) sl = slot[s_ent];
        s_slot = sl;
    }
    __syncthreads();
    const int sl = s_slot;

    for (int r = t; r < R; r += blockDim.x)
        s_cnt[r] = (sl >= 0) ? cnt[(long long)sl * R + r] : 0.0f;
    __syncthreads();

    // deg = sum_r cnt[r]
    float p = 0.0f;
    for (int r = t; r < R; r += blockDim.x) p += s_cnt[r];
    s_part[t] = p;
    __syncthreads();
    for (int o = 128; o > 0; o >>= 1) {
        if (t < o) s_part[t] += s_part[t + o];
        __syncthreads();
    }
    if (t == 0) s_deg = s_part[0];
    __syncthreads();
    const float deg = s_deg;

    // num[d] = sum_r cnt[r] * emb[b, r, d]  (coalesced streaming of emb row block)
    const int d  = t % D;            // D == 64
    const int rl = t / D;            // 0..3
    const int G  = blockDim.x / D;   // 4
    const float* __restrict__ eb = emb + (long long)b * R * D;
    float acc = 0.0f;
    for (int r = rl; r < R; r += G) acc += s_cnt[r] * eb[(long long)r * D + d];
    __syncthreads();
    s_part[t] = acc;
    __syncthreads();

    const float fE   = (float)((E > 0) ? E : 1);
    const bool  vent = (s_ent >= 0 && s_ent < s_N);
    const bool  vrel = (s_qr >= 0 && s_qr < R);
    float* __restrict__ frow = feat + (long long)b * KF;

    if (t < D) {
        float num = 0.0f;
        for (int g = 0; g < G; ++g) num += s_part[g * D + t];
        float ee = (vent && deg > 0.0f) ? (num / fmaxf(deg, 1.0f)) : 0.0f;
        float re = vrel ? eb[(long long)s_qr * D + t] : 0.0f;
        frow[t]     = re;   // rel_emb
        frow[D + t] = ee;   // entity_emb
    }
    if (t == 0) {
        int   qc   = vrel ? s_qr : 0;
        float rf   = vrel ? fminf(relc[qc] / fE, 1.0f) : 0.0f;
        float dn   = fminf((vent ? deg : 0.0f) / fE, 1.0f);
        float Nf   = (float)s_N;
        float dens = fminf((float)E / fmaxf(Nf * Nf, 1.0f), 1.0f);
        frow[2 * D + 0] = rf;
        frow[2 * D + 1] = dn;
        frow[2 * D + 2] = rf;
        frow[2 * D + 3] = dens;
    }
}

// ---------------------------------------------------------------------------
// Kernel 5: O = relu(A[M,K] @ W[K,N] + bias), f32 WMMA 16x16x4.
// One wave (32 thr) per 16x16 output tile. K multiple of 4, M,N multiples of 16.
// A frag (16x4): lane L -> M = L%16, VGPR v holds K = k0 + 2*(L/16) + v
// B frag (4x16): lane L -> N = L%16, VGPR v holds K = k0 + 2*(L/16) + v
// C/D (16x16):   lane L -> N = L%16, VGPR v holds M = v + 8*(L/16)
// ---------------------------------------------------------------------------
__global__ void k_gemm_relu_wmma(const float* __restrict__ A, int lda,
                                 const float* __restrict__ W,
                                 const float* __restrict__ bias,
                                 float* __restrict__ O,
                                 int M, int K, int N) {
    const int ntn  = N >> 4;
    const int mt   = blockIdx.x / ntn;
    const int nt   = blockIdx.x % ntn;
    const int lane = threadIdx.x;
    const int l16  = lane & 15;
    const int half = lane >> 4;

    const float* __restrict__ arow = A + (long long)(mt * 16 + l16) * lda;
    const float* __restrict__ wcol = W + (nt * 16 + l16);

    v8f c = {};
    for (int k0 = 0; k0 < K; k0 += 4) {
        const int ka = k0 + 2 * half;
        v2f a, bb;
        a.x  = arow[ka];
        a.y  = arow[ka + 1];
        bb.x = wcol[(long long)ka * N];
        bb.y = wcol[(long long)(ka + 1) * N];
        c = __builtin_amdgcn_wmma_f32_16x16x4_f32(
                /*neg_a=*/false, a, /*neg_b=*/false, bb,
                /*c_mod=*/(short)0, c, /*reuse_a=*/false, /*reuse_b=*/false);
    }

    const int n  = nt * 16 + l16;
    const float bv = bias[n];
#pragma unroll
    for (int v = 0; v < 8; ++v) {
        int m = mt * 16 + v + 8 * half;
        float val = c[v] + bv;
        O[(long long)m * N + n] = fmaxf(val, 0.0f);
    }
}

// ---------------------------------------------------------------------------
// Kernel 6: gate = sigmoid(H3[M,16] @ W4[16,1] + b4)
// ---------------------------------------------------------------------------
__global__ void k_head(const float* __restrict__ H3, const float* __restrict__ W4,
                       const float* __restrict__ b4, float* __restrict__ out,
                       int M, int Kh) {
    int b = blockIdx.x * blockDim.x + threadIdx.x;
    if (b >= M) return;
    float acc = b4[0];
    const float* h = H3 + (long long)b * Kh;
    for (int j = 0; j < Kh; ++j) acc += h[j] * W4[j];
    out[b] = 1.0f / (1.0f + expf(-acc));
}

// ---------------------------------------------------------------------------
extern "C" void kernel_launch(void* const* d_in, const int* in_sizes, int n_in,
                              void* d_out, int out_size, void* d_ws, size_t ws_size,
                              hipStream_t stream) {
    const float* emb   = (const float*)d_in[0];
    const int*   qrels = (const int*)d_in[1];
    const int*   qents = (const int*)d_in[2];
    const int*   eidx  = (const int*)d_in[3];
    const int*   etyp  = (const int*)d_in[4];
    const int*   d_nn  = (const int*)d_in[5];   // num_nodes (device scalar)
    const float* W1 = (const float*)d_in[7];
    const float* b1 = (const float*)d_in[8];
    const float* W2 = (const float*)d_in[9];
    const float* b2 = (const float*)d_in[10];
    const float* W3 = (const float*)d_in[11];
    const float* b3 = (const float*)d_in[12];
    const float* W4 = (const float*)d_in[13];
    const float* b4 = (const float*)d_in[14];

    const int       B  = in_sizes[1];
    const long long E  = (long long)in_sizes[4];
    const int       D  = in_sizes[8];                                  // 64
    const int       R  = (int)((long long)in_sizes[0] / ((long long)B * D)); // 512
    const int       D2 = D / 2, D4 = D / 4;
    const int       KF = 2 * D + 4;                                    // 132

    const int* esrc = eidx;
    const int* edst = eidx + E;

    auto al = [](size_t x) { return (x + 255) & ~(size_t)255; };
    size_t off = 0;
    float* cnt  = (float*)((char*)d_ws + off); off = al(off + (size_t)B * R * 4);
    float* relc = (float*)((char*)d_ws + off); off = al(off + (size_t)R * 4);
    float* feat = (float*)((char*)d_ws + off); off = al(off + (size_t)B * KF * 4);
    float* H1   = (float*)((char*)d_ws + off); off = al(off + (size_t)B * D * 4);
    float* H2   = (float*)((char*)d_ws + off); off = al(off + (size_t)B * D2 * 4);
    float* H3   = (float*)((char*)d_ws + off); off = al(off + (size_t)B * D4 * 4);
    int*   slot = (int*)((char*)d_ws + off);   // N ints, tail of workspace

    k_init <<<1024, 256, 0, stream>>>(cnt, (long long)B * R, relc, R, slot, d_nn);
    k_mark <<<(B + 255) / 256, 256, 0, stream>>>(qents, B, d_nn, slot);
    k_edges<<<1536, 256, 0, stream>>>(esrc, edst, etyp, E, R, slot, d_nn, cnt, relc);
    k_feat <<<B, 256, 0, stream>>>(emb, qrels, qents, d_nn, E, R, D, slot, cnt, relc, feat, KF);

    k_gemm_relu_wmma<<<(B / 16) * (D  / 16), 32, 0, stream>>>(feat, KF, W1, b1, H1, B, KF, D);
    k_gemm_relu_wmma<<<(B / 16) * (D2 / 16), 32, 0, stream>>>(H1,  D,  W2, b2, H2, B, D,  D2);
    k_gemm_relu_wmma<<<(B / 16) * (D4 / 16), 32, 0, stream>>>(H2,  D2, W3, b3, H3, B, D2, D4);

    k_head <<<(B + 255) / 256, 256, 0, stream>>>(H3, W4, b4, (float*)d_out, B, D4);
}